// MultiHeadAttention_53223234732610
// MI455X (gfx1250) — compile-verified
//
#include <hip/hip_runtime.h>
#include <hip/hip_bf16.h>

typedef __attribute__((ext_vector_type(2))) float v2f;
typedef __attribute__((ext_vector_type(4))) float v4f;
typedef __attribute__((ext_vector_type(8))) float v8f;
typedef int v4i __attribute__((vector_size(4 * sizeof(int))));   // matches async builtin param

#define N_USERS 16384
#define EMB 32
#define KC 64                    // K-chunk staged in LDS per iteration
#define LDS_A_STRIDE (KC + 4)    // 68 floats: float4-aligned rows, 4-bank shift per row
#define WAVES 4
#define MB (WAVES * 16)          // 64 output rows per block
#define THREADS (WAVES * 32)     // 128 threads, wave32

#if __has_builtin(__builtin_amdgcn_global_load_async_to_lds_b128)
#define ASYNC_COPY 1
#else
#define ASYNC_COPY 0
#endif

__device__ __forceinline__ void async_copy_wait() {
#if ASYNC_COPY
#if __has_builtin(__builtin_amdgcn_s_wait_asynccnt)
    __builtin_amdgcn_s_wait_asynccnt(0);
#else
    asm volatile("s_wait_asynccnt 0" ::: "memory");
#endif
#endif
}

// 16-byte global -> LDS copy: async DMA (ASYNCcnt) if available, else ld+st.
__device__ __forceinline__ void copy16(const float* __restrict__ g, float* __restrict__ l) {
#if ASYNC_COPY
    __builtin_amdgcn_global_load_async_to_lds_b128(
        (__attribute__((address_space(1))) v4i*)(void*)g,
        (__attribute__((address_space(3))) v4i*)(void*)l, 0, 0);
#else
    *(v4f*)l = *(const v4f*)g;
#endif
}

// Pre-swizzle preference_emb into WMMA B-fragment order:
// Bsw (as float2)[(s*2 + t)*32 + lane] = { B[4s + 2*half][t*16 + lrow],
//                                          B[4s + 2*half + 1][t*16 + lrow] }
// (s = global k-step of 4, t = 16-wide N tile, lane = WMMA lane).
// Chunk offsets in Bsw coincide with row-major chunk offsets, so staging is a straight copy.
__global__ void swizzle_b_kernel(const float* __restrict__ pref, float* __restrict__ bsw) {
    const int i    = blockIdx.x * blockDim.x + threadIdx.x;   // float2 index
    const int lane = i & 31;
    const int t    = (i >> 5) & 1;
    const int s    = i >> 6;
    const int half = lane >> 4;
    const int lrow = lane & 15;
    const int n = t * 16 + lrow;
    const int k = 4 * s + 2 * half;
    v2f b;
    b.x = pref[(size_t)k * EMB + n];
    b.y = pref[(size_t)(k + 1) * EMB + n];
    *(v2f*)&bsw[(size_t)i * 2] = b;
}

// Stage one K-chunk: A tile (MB x KC, padded) and swizzled-B tile (contiguous 8 KB).
__device__ __forceinline__ void stage_chunk(const float* __restrict__ attn,
                                            const float* __restrict__ bsw,
                                            int rowBase, int k0,
                                            float* __restrict__ bufA,
                                            float* __restrict__ bufB, int tid) {
    // A: MB*KC/4 = 1024 float4, 8 per thread, 16 float4 per row (coalesced per 16 lanes)
    #pragma unroll
    for (int j = 0; j < (MB * (KC / 4)) / THREADS; ++j) {
        const int i  = tid + j * THREADS;
        const int r  = i >> 4;
        const int c4 = i & 15;
        copy16(attn + (size_t)(rowBase + r) * N_USERS + (size_t)k0 + (size_t)(c4 * 4),
               &bufA[r * LDS_A_STRIDE + c4 * 4]);
    }
    // B (swizzled): one contiguous 8 KB region starting at float offset k0*EMB
    #pragma unroll
    for (int j = 0; j < (KC * EMB / 4) / THREADS; ++j) {
        const int i = tid + j * THREADS;
        copy16(bsw + (size_t)k0 * EMB + (size_t)(i * 4), &bufB[i * 4]);
    }
#if !ASYNC_COPY
    if (k0 + 2 * KC < N_USERS)
        __builtin_prefetch(attn + (size_t)(rowBase + (tid >> 1)) * N_USERS + k0 + 2 * KC, 0, 1);
#endif
}

// out_tmp[N,32] = attention[N,N] @ pref[N,32]  (fp32 WMMA 16x16x4, double-buffered LDS)
__global__ __launch_bounds__(THREADS)
void gemm_attn_wmma(const float* __restrict__ attn,
                    const float* __restrict__ bsw,
                    float* __restrict__ out_tmp) {
    __shared__ float ldsA[2][MB * LDS_A_STRIDE];   // 2 x 17408 B
    __shared__ float ldsB[2][KC * EMB];            // 2 x  8192 B

    const int tid   = threadIdx.x;
    const int wave  = tid >> 5;
    const int lane  = tid & 31;
    const int half  = lane >> 4;     // 0: lanes 0-15, 1: lanes 16-31
    const int lrow  = lane & 15;
    const int rowBase = blockIdx.x * MB;
    const int waveRow = wave * 16;

    v8f acc0 = {};   // output cols 0..15
    v8f acc1 = {};   // output cols 16..31

    // prologue: stage chunk 0 into buffer 0
    stage_chunk(attn, bsw, rowBase, 0, ldsA[0], ldsB[0], tid);
    async_copy_wait();
    __syncthreads();

    int buf = 0;
    for (int k0 = 0; k0 < N_USERS; k0 += KC) {
        const int nxt = buf ^ 1;
        // kick off next chunk's DMA before computing on the current one
        if (k0 + KC < N_USERS)
            stage_chunk(attn, bsw, rowBase, k0 + KC, ldsA[nxt], ldsB[nxt], tid);

        const float* bufA = ldsA[buf];
        const float* bufB = ldsB[buf];

        #pragma unroll
        for (int kk = 0; kk < KC; kk += 4) {
            // A fragment (16x4): lanes 0-15 rows 0-15 K={kk,kk+1}; lanes 16-31 K={kk+2,kk+3}
            v2f a = *(const v2f*)&bufA[(waveRow + lrow) * LDS_A_STRIDE + kk + 2 * half];

            // B fragments: pre-swizzled, one b64 per tile, lane-indexed (bank-conflict-free)
            const int step = kk >> 2;
            v2f b0 = *(const v2f*)&bufB[(step * 64 + lane) * 2];        // cols 0..15
            v2f b1 = *(const v2f*)&bufB[(step * 64 + 32 + lane) * 2];   // cols 16..31

            acc0 = __builtin_amdgcn_wmma_f32_16x16x4_f32(
                       false, a, false, b0, (short)0, acc0, false, false);
            acc1 = __builtin_amdgcn_wmma_f32_16x16x4_f32(
                       false, a, false, b1, (short)0, acc1, false, false);
        }

        async_copy_wait();   // next chunk's DMA landed in LDS
        __syncthreads();     // everyone done reading current buffer
        buf = nxt;
    }

    // C/D layout: VGPR i -> lanes 0-15: (M=i, N=lane); lanes 16-31: (M=8+i, N=lane-16)
    const int outRow = rowBase + waveRow + half * 8;
    #pragma unroll
    for (int i = 0; i < 8; ++i) {
        out_tmp[(size_t)(outRow + i) * EMB + lrow]      = acc0[i];
        out_tmp[(size_t)(outRow + i) * EMB + 16 + lrow] = acc1[i];
    }
}

__global__ void zero_out_kernel(float* __restrict__ out, int n) {
    const int i = blockIdx.x * blockDim.x + threadIdx.x;
    if (i < n) out[i] = 0.0f;
}

// out[dst[e]] += tmp[src[e]]  — one wave per edge, lane = embedding dim
__global__ void edge_scatter_kernel(const float* __restrict__ tmp,
                                    const int* __restrict__ src,
                                    const int* __restrict__ dst,
                                    float* __restrict__ out, int n_edges) {
    const long long idx = (long long)blockIdx.x * blockDim.x + threadIdx.x;
    const long long e = idx >> 5;
    const int d = (int)(idx & 31);
    if (e < n_edges) {
        const int s = src[e];
        const int t = dst[e];
        const float v = tmp[(size_t)s * EMB + d];
        unsafeAtomicAdd(&out[(size_t)t * EMB + d], v);   // native global_atomic_add_f32
    }
}

extern "C" void kernel_launch(void* const* d_in, const int* in_sizes, int n_in,
                              void* d_out, int out_size, void* d_ws, size_t ws_size,
                              hipStream_t stream) {
    // inputs: [0] social_emb (unused), [1] preference_emb, [2] attention_scores, [3] uu_graph
    const float* pref = (const float*)d_in[1];
    const float* attn = (const float*)d_in[2];
    const int*   uu   = (const int*)d_in[3];
    const int n_edges = in_sizes[3] / 2;
    const int* src = uu;
    const int* dst = uu + n_edges;

    float* out = (float*)d_out;
    float* tmp = (float*)d_ws;                       // 2 MB
    float* bsw = (float*)d_ws + (size_t)N_USERS * EMB; // 2 MB swizzled B

    // 0) swizzle B into WMMA fragment order (N_USERS/4 steps * 2 tiles * 32 lanes float2)
    const int n_frag2 = (N_USERS / 4) * 2 * 32;
    swizzle_b_kernel<<<n_frag2 / 256, 256, 0, stream>>>(pref, bsw);

    // 1) tmp = attention @ pref  (fp32 WMMA)
    gemm_attn_wmma<<<N_USERS / MB, THREADS, 0, stream>>>(attn, bsw, tmp);

    // 2) out = 0
    const int n_out = N_USERS * EMB;
    zero_out_kernel<<<(n_out + 255) / 256, 256, 0, stream>>>(out, n_out);

    // 3) scatter-add over edges
    const long long total = (long long)n_edges * 32;
    const int blocks = (int)((total + 255) / 256);
    edge_scatter_kernel<<<blocks, 256, 0, stream>>>(tmp, src, dst, out, n_edges);
}